// Net_53747220742790
// MI455X (gfx1250) — compile-verified
//
#include <hip/hip_runtime.h>
#include <hip/hip_bf16.h>

typedef __attribute__((ext_vector_type(2))) float v2f;
typedef __attribute__((ext_vector_type(8))) float v8f;

// ---------------------------------------------------------------------------
// deg[i] = 1.0 (self-loop weight); later accumulated with edge weights.
// ---------------------------------------------------------------------------
__global__ void k_init_deg(float* __restrict__ deg, int N) {
    int i = blockIdx.x * blockDim.x + threadIdx.x;
    if (i < N) deg[i] = 1.0f;
}

// deg[dst] += w  over all edges
__global__ void k_deg_edges(const int* __restrict__ ei, const float* __restrict__ w,
                            float* __restrict__ deg, int E) {
    int e = blockIdx.x * blockDim.x + threadIdx.x;
    if (e < E) {
        int d = ei[E + e];              // row 1 of (2,E) edge_index
        atomicAdd(&deg[d], w[e]);
    }
}

// dinv = rsqrt(deg) (deg>0 always holds due to self-loop), in place.
// agg1[i] init = selfloop contribution = dinv^2 * x[i]
__global__ void k_dinv_init1(const float* __restrict__ x, float* __restrict__ deg_dinv,
                             float* __restrict__ agg1, int N) {
    int i = blockIdx.x * blockDim.x + threadIdx.x;
    if (i < N) {
        float d  = deg_dinv[i];
        float di = (d > 0.0f) ? rsqrtf(d) : 0.0f;
        deg_dinv[i] = di;
        agg1[i] = di * di * x[i];
    }
}

// Layer-1 edge pass (scalar messages): norm[e] = dinv[s]*w*dinv[d];
// agg1[d] += norm * x[s]
__global__ void k_edge1(const int* __restrict__ ei, const float* __restrict__ w,
                        const float* __restrict__ dinv, const float* __restrict__ x,
                        float* __restrict__ norm, float* __restrict__ agg1, int E) {
    int e = blockIdx.x * blockDim.x + threadIdx.x;
    if (e < E) {
        int s = ei[e];
        int d = ei[E + e];
        float nm = dinv[s] * w[e] * dinv[d];
        norm[e] = nm;
        atomicAdd(&agg1[d], nm * x[s]);
    }
}

// h[i][f] = relu(agg1[i]*W1[f] + b1[f]);  one thread per (node, feature)
__global__ void k_h1(const float* __restrict__ agg1, const float* __restrict__ W1,
                     const float* __restrict__ b1, float* __restrict__ h, int N) {
    int t = blockIdx.x * blockDim.x + threadIdx.x;
    if (t < N * 16) {
        int i = t >> 4, f = t & 15;
        float v = fmaf(agg1[i], W1[f], b1[f]);
        h[t] = v > 0.0f ? v : 0.0f;
    }
}

// ---------------------------------------------------------------------------
// Layer-2 transform via WMMA:  z = h @ W2pad  (W2 16x2 zero-padded to 16x16)
// One wave computes a 16-node x 16-col tile, K=16 as 4 chained K=4 WMMAs.
// Fused: agg2[i][c] init = dinv[i]^2 * z[i][c]  (self-loop term of layer 2).
// ---------------------------------------------------------------------------
__global__ void k_layer2_wmma(const float* __restrict__ h, const float* __restrict__ W2,
                              const float* __restrict__ dinv, float* __restrict__ z,
                              float* __restrict__ agg2, int N) {
    __shared__ float w2p[16 * 16];      // [k][n], cols >= 2 are zero
    int t = threadIdx.x;                // blockDim.x == 256
    {
        int k = t >> 4, n = t & 15;
        w2p[t] = (n < 2) ? W2[k * 2 + n] : 0.0f;
    }
    __syncthreads();

    int wave = t >> 5;                  // 8 waves per block
    int lane = t & 31;
    int tile = blockIdx.x * 8 + wave;
    int m0   = tile * 16;
    if (m0 >= N) return;                // wave-uniform exit

    int row   = lane & 15;              // M (for A/D) or N (for B)
    int khalf = (lane >> 4) * 2;        // lanes 16-31 hold K+2,K+3

    // clamp load row for partial tiles (stores are guarded below)
    int anode = m0 + row;
    int aload = anode < N ? anode : N - 1;

    v8f acc = {};
#pragma unroll
    for (int tstep = 0; tstep < 4; ++tstep) {
        int kb = tstep * 4 + khalf;
        v2f a, b;
        // A 16x4 f32 layout: lane L<16 -> {K=kb, K=kb+1} of row L; L>=16 -> {kb+2, kb+3}
        a.x = h[aload * 16 + kb];
        a.y = h[aload * 16 + kb + 1];
        // B 4x16 f32 layout (mirror of A): lane L -> column (L&15) of rows kb, kb+1
        b.x = w2p[kb * 16 + row];
        b.y = w2p[(kb + 1) * 16 + row];
        acc = __builtin_amdgcn_wmma_f32_16x16x4_f32(
            /*neg_a=*/false, a, /*neg_b=*/false, b,
            /*c_mod=*/(short)0, acc, /*reuse_a=*/false, /*reuse_b=*/false);
    }

    // D layout: VGPR r -> row (m0 + (lane<16 ? r : 8+r)), col (lane & 15).
    int col = row;
    if (col < 2) {
        int mofs = (lane >> 4) * 8;
#pragma unroll
        for (int r = 0; r < 8; ++r) {
            int node = m0 + mofs + r;
            if (node < N) {
                float zv = acc[r];
                float di = dinv[node];
                z[node * 2 + col]    = zv;
                agg2[node * 2 + col] = di * di * zv;   // self-loop init
            }
        }
    }
}

// Layer-2 edge pass (2-wide messages): agg2[d][c] += norm[e] * z[s][c]
__global__ void k_edge2(const int* __restrict__ ei, const float* __restrict__ norm,
                        const float* __restrict__ z, float* __restrict__ agg2, int E) {
    int e = blockIdx.x * blockDim.x + threadIdx.x;
    if (e < E) {
        int s = ei[e];
        int d = ei[E + e];
        float nm = norm[e];
        float z0 = z[2 * s], z1 = z[2 * s + 1];
        atomicAdd(&agg2[2 * d],     nm * z0);
        atomicAdd(&agg2[2 * d + 1], nm * z1);
    }
}

// out = log_softmax(agg2 + b2) over 2 classes (numerically stable)
__global__ void k_out(const float* __restrict__ agg2, const float* __restrict__ b2,
                      float* __restrict__ out, int N) {
    int i = blockIdx.x * blockDim.x + threadIdx.x;
    if (i < N) {
        float a0 = agg2[2 * i]     + b2[0];
        float a1 = agg2[2 * i + 1] + b2[1];
        float m  = fmaxf(a0, a1);
        float lse = m + logf(expf(a0 - m) + expf(a1 - m));
        out[2 * i]     = a0 - lse;
        out[2 * i + 1] = a1 - lse;
    }
}

extern "C" void kernel_launch(void* const* d_in, const int* in_sizes, int n_in,
                              void* d_out, int out_size, void* d_ws, size_t ws_size,
                              hipStream_t stream) {
    const float* x   = (const float*)d_in[0];   // [N,1]
    const int*   ei  = (const int*)  d_in[1];   // [2,E]
    const float* ew  = (const float*)d_in[2];   // [E]
    const float* W1  = (const float*)d_in[3];   // [1,16]
    const float* b1  = (const float*)d_in[4];   // [16]
    const float* W2  = (const float*)d_in[5];   // [16,2]
    const float* b2  = (const float*)d_in[6];   // [2]
    float*       out = (float*)d_out;           // [N,2]

    const int N = in_sizes[0];
    const int E = in_sizes[2];

    // workspace layout (floats): dinv[N] | agg1[N] | norm[E] | h[16N] | z[2N] | agg2[2N]
    float* wsf   = (float*)d_ws;
    float* dinv  = wsf;
    float* agg1  = wsf + (size_t)N;
    float* norm  = wsf + (size_t)2 * N;
    float* h     = norm + (size_t)E;
    float* z     = h + (size_t)16 * N;
    float* agg2  = z + (size_t)2 * N;

    const int B = 256;
    const int gN  = (N + B - 1) / B;
    const int gE  = (E + B - 1) / B;
    const int gNF = (N * 16 + B - 1) / B;
    const int tiles = (N + 15) / 16;
    const int gW  = (tiles + 7) / 8;    // 8 waves (tiles) per 256-thread block

    k_init_deg   <<<gN,  B, 0, stream>>>(dinv, N);               // dinv holds deg first
    k_deg_edges  <<<gE,  B, 0, stream>>>(ei, ew, dinv, E);
    k_dinv_init1 <<<gN,  B, 0, stream>>>(x, dinv, agg1, N);
    k_edge1      <<<gE,  B, 0, stream>>>(ei, ew, dinv, x, norm, agg1, E);
    k_h1         <<<gNF, B, 0, stream>>>(agg1, W1, b1, h, N);
    k_layer2_wmma<<<gW,  B, 0, stream>>>(h, W2, dinv, z, agg2, N);
    k_edge2      <<<gE,  B, 0, stream>>>(ei, norm, z, agg2, E);
    k_out        <<<gN,  B, 0, stream>>>(agg2, b2, out, N);
}